// CayleyConv_14577119002998
// MI455X (gfx1250) — compile-verified
//
#include <hip/hip_runtime.h>

typedef __attribute__((ext_vector_type(2))) float v2f;
typedef __attribute__((ext_vector_type(8))) float v8f;
// 4-float vector with 4-byte alignment: lets us emit b128 loads/stores at
// DWORD-aligned (not 16B-aligned) addresses; legal in DWORD alignment mode.
typedef float f4u __attribute__((ext_vector_type(4), aligned(4)));

#define BATCH 16
#define OCH   16
#define CIN   16
#define HH    128
#define WW    128
// output strides (floats): [b][o][130][130][3][3]
#define OSTR_O  (130*130*9)     // 152100
#define OSTR_R  (130*9)         // 1170
#define OSTR_B  (OCH*OSTR_O)    // 2433600
// x strides (floats): [b][i][128][128][3][3]
#define XSTR_I  (HH*WW*9)
#define XSTR_B  (CIN*XSTR_I)

// ---------------------------------------------------------------------------
// Kernel 1: build G[o][i][a][b][p][q]  (Cayley transform, exact ref formulas)
// one thread per (o,i,tap) -> 16*16*9 = 2304 threads
// ---------------------------------------------------------------------------
__global__ void k_build_G(const float* __restrict__ g, float* __restrict__ Gws) {
    int tid = blockIdx.x * blockDim.x + threadIdx.x;
    if (tid >= OCH * CIN * 9) return;
    int tap = tid % 9;            // a*3 + b
    int oi  = tid / 9;            // o*CIN + i
    int tr  = tap / 3, tc = tap % 3;
    const int idxm[9] = {0, 1, 2, 3, 4, 5, 6, 7, 4};
    int kk = idxm[tap];
    const float* gp = g + (size_t)oi * 24 + kk * 3;   // g[o][i][kk][0..2]
    float av = gp[0], bv = gp[1], cv = gp[2];
    float A0[9] = {0.f, av, bv, -av, 0.f, cv, -bv, -cv, 0.f};
    float Xm[9], Idm[9];
#pragma unroll
    for (int e = 0; e < 9; ++e) {
        float id = (e == 0 || e == 4 || e == 8) ? 1.f : 0.f;
        if (tr == 1 && tc == 1) { Xm[e] = id; Idm[e] = id; }
        else                    { Xm[e] = A0[e]; Idm[e] = id - A0[e]; }
    }
    float b00 = Idm[0], b01 = Idm[1], b02 = Idm[2];
    float b10 = Idm[3], b11 = Idm[4], b12 = Idm[5];
    float b20 = Idm[6], b21 = Idm[7], b22 = Idm[8];
    float det = b00*(b11*b22 - b12*b21) - b01*(b10*b22 - b12*b20)
              + b02*(b10*b21 - b11*b20);
    float d = det + 1e-7f;
    float inv[9];
    inv[0] = (b11*b22 - b12*b21) / d;  inv[1] = (b02*b21 - b01*b22) / d;
    inv[2] = (b01*b12 - b02*b11) / d;  inv[3] = (b12*b20 - b10*b22) / d;
    inv[4] = (b00*b22 - b02*b20) / d;  inv[5] = (b02*b10 - b00*b12) / d;
    inv[6] = (b10*b21 - b11*b20) / d;  inv[7] = (b01*b20 - b00*b21) / d;
    inv[8] = (b00*b11 - b01*b10) / d;
    float M2[9];
#pragma unroll
    for (int e = 0; e < 9; ++e)
        M2[e] = Xm[e] + ((e == 0 || e == 4 || e == 8) ? 1.f : 0.f);
    float* Gp = Gws + (size_t)tid * 9;
#pragma unroll
    for (int p = 0; p < 3; ++p)
#pragma unroll
        for (int q = 0; q < 3; ++q)
            Gp[p*3 + q] = inv[p*3+0]*M2[0+q] + inv[p*3+1]*M2[3+q] + inv[p*3+2]*M2[6+q];
}

// ---------------------------------------------------------------------------
// Kernel 2: interior weight matrix Wm[144][48]
// row m = o*9 + p*3 + t ; col k = tap*9 + q*3 + s (tap 0 = identity, 1..4 = Gl
// at taps (1,0),(0,2),(0,1),(0,0)); cols 45..47 zero pad.
// ---------------------------------------------------------------------------
__global__ void k_build_Wm(const float* __restrict__ Gws, float* __restrict__ Wm) {
    int tid = blockIdx.x * blockDim.x + threadIdx.x;
    if (tid >= 144 * 48) return;
    int m = tid / 48, k = tid % 48;
    int o = m / 9, p = (m % 9) / 3, t = m % 3;
    float v = 0.f;
    if (k < 45) {
        int tap = k / 9, q = (k % 9) / 3, s = k % 3;
        if (tap == 0) {
            v = (p == q && t == s) ? 1.f : 0.f;
        } else {
            const int ta[4] = {1, 0, 0, 0};
            const int tb[4] = {0, 2, 1, 0};
            int a = ta[tap-1], b = tb[tap-1];
            const float* Gl = Gws + (((size_t)(o*CIN + (CIN-1))*3 + a)*3 + b) * 9;
            v = Gl[p*3 + q] * Gl[t*3 + s];
        }
    }
    Wm[tid] = v;
}

// ---------------------------------------------------------------------------
// Kernel 3: interior via V_WMMA_F32_16X16X4_F32.
// One wave = 16 positions (same h, 16 consecutive w) x all 144 output rows.
// 9 row-blocks x 12 K-steps of wmma; result staged in LDS [o][pos][pt] and
// written back as contiguous b128 stores (576 B run per output channel).
// All lanes active (EXEC all 1s as WMMA requires).
// ---------------------------------------------------------------------------
__global__ __launch_bounds__(128) void k_interior(const float* __restrict__ x,
                                                  const float* __restrict__ Wm,
                                                  float* __restrict__ out) {
    __shared__ float lds[4 * 2304];                      // 36 KB, 4 waves/block
    int wslot = threadIdx.x >> 5;
    int wid   = (blockIdx.x * 128 + threadIdx.x) >> 5;   // 0..16383
    int lane  = threadIdx.x & 31;
    int b     = wid >> 10;                               // / (128*128/16)
    int tile  = wid & 1023;
    int n0    = tile << 4;
    int h     = n0 >> 7;                                 // row (0..127)
    int w0    = n0 & 127;
    int col   = lane & 15;
    int hi    = lane >> 4;                               // half-wave select
    int w     = w0 + col;

    const float* xl = x + (size_t)(b*CIN + (CIN-1)) * XSTR_I;  // last channel

    // B fragments: lane holds X[K][n], K = 4*ks + 2*hi + j, n = col
    v2f Bv[12];
#pragma unroll
    for (int ks = 0; ks < 12; ++ks) {
#pragma unroll
        for (int j = 0; j < 2; ++j) {
            int k = 4*ks + 2*hi + j;
            float val = 0.f;
            if (k < 45) {
                int tap = k / 9, e = k % 9;
                int hs = h, ws = w;
                bool ok = true;
                if      (tap == 1) { ws = w + 1; ok = (ws < WW); }
                else if (tap == 2) { hs = h + 1; ws = w - 1; ok = (hs < HH) && (ws >= 0); }
                else if (tap == 3) { hs = h + 1; ok = (hs < HH); }
                else if (tap == 4) { hs = h + 1; ws = w + 1; ok = (hs < HH) && (ws < WW); }
                if (ok) val = xl[(size_t)(hs*WW + ws)*9 + e];
            }
            Bv[ks][j] = val;
        }
    }

    float* wbuf = lds + wslot * 2304;

#pragma unroll
    for (int rb = 0; rb < 9; ++rb) {
        v8f acc = {};
        const float* Arow = Wm + (size_t)(16*rb + col) * 48 + 2*hi;
#pragma unroll
        for (int ks = 0; ks < 12; ++ks) {
            v2f a;
            a[0] = Arow[4*ks + 0];
            a[1] = Arow[4*ks + 1];
            acc = __builtin_amdgcn_wmma_f32_16x16x4_f32(
                false, a, false, Bv[ks], (short)0, acc, false, false);
        }
        // C layout: lane col = lane%16, rows = 16*rb + 8*hi + r
        // stage into LDS as [o][pos][pt]
#pragma unroll
        for (int r = 0; r < 8; ++r) {
            int row = 16*rb + 8*hi + r;     // 0..143 = o*9 + p*3 + t
            int oo  = row / 9;
            int pt  = row % 9;
            wbuf[oo*144 + col*9 + pt] = acc[r];
        }
    }

    __builtin_amdgcn_wave_barrier();   // same-wave LDS reuse; scheduling fence

    // Coalesced writeback: 2304 floats = 18 wave-wide b128 stores.
    // Per o: contiguous 144-float (576 B) run at out[b][o][h+1][w0+1][0][0].
    // 144 % 4 == 0, so no aligned 4-float group crosses an o-chunk boundary.
    float* outb = out + (size_t)b * OSTR_B;
    size_t rowBase = (size_t)(h + 1) * OSTR_R + (size_t)(w0 + 1) * 9;
    const f4u* wb4 = (const f4u*)wbuf;
#pragma unroll
    for (int i = 0; i < 18; ++i) {
        int f   = i * 128 + lane * 4;       // flat index, multiple of 4
        int oo  = f / 144;
        int rem = f - oo * 144;
        f4u v   = wb4[f >> 2];
        *(f4u*)(outb + (size_t)oo * OSTR_O + rowBase + rem) = v;
    }
}

// ---------------------------------------------------------------------------
// Kernel 4: rim of the 130x130 frame (516 positions per (b,o)) keeps the
// accumulated 9-tap, 16-channel sum. One thread per (b,o,rim position).
// ---------------------------------------------------------------------------
__global__ void k_border(const float* __restrict__ x, const float* __restrict__ Gws,
                         float* __restrict__ out) {
    int tid = blockIdx.x * blockDim.x + threadIdx.x;   // 132096 total
    int rim = tid % 516;
    int bo  = tid / 516;
    int b = bo >> 4, o = bo & 15;
    int r, c;
    if      (rim < 130) { r = 0;   c = rim; }
    else if (rim < 260) { r = 129; c = rim - 130; }
    else if (rim < 388) { c = 0;   r = rim - 259; }    // r = 1..128
    else                { c = 129; r = rim - 387; }    // r = 1..128

    float acc[9];
#pragma unroll
    for (int e = 0; e < 9; ++e) acc[e] = 0.f;

#pragma unroll
    for (int a = 0; a < 3; ++a) {
        int hh = r - a;
        if (hh < 0 || hh >= HH) continue;
#pragma unroll
        for (int bb = 0; bb < 3; ++bb) {
            int ww = c - bb;
            if (ww < 0 || ww >= WW) continue;
            for (int i = 0; i < CIN; ++i) {
                const float* Gm = Gws + ((size_t)((o*CIN + i)*9 + a*3 + bb)) * 9;
                const float* xv = x   + ((size_t)((b*CIN + i)*HH + hh)*WW + ww) * 9;
                float P[9];
#pragma unroll
                for (int p = 0; p < 3; ++p)
#pragma unroll
                    for (int s = 0; s < 3; ++s)
                        P[p*3+s] = Gm[p*3+0]*xv[0+s] + Gm[p*3+1]*xv[3+s] + Gm[p*3+2]*xv[6+s];
#pragma unroll
                for (int p = 0; p < 3; ++p)
#pragma unroll
                    for (int t = 0; t < 3; ++t)
                        acc[p*3+t] += P[p*3+0]*Gm[t*3+0] + P[p*3+1]*Gm[t*3+1] + P[p*3+2]*Gm[t*3+2];
            }
        }
    }
    float* op = out + (size_t)b * OSTR_B + (size_t)o * OSTR_O
              + (size_t)r * OSTR_R + (size_t)c * 9;
#pragma unroll
    for (int e = 0; e < 9; ++e) op[e] = acc[e];
}

// ---------------------------------------------------------------------------
extern "C" void kernel_launch(void* const* d_in, const int* in_sizes, int n_in,
                              void* d_out, int out_size, void* d_ws, size_t ws_size,
                              hipStream_t stream) {
    const float* x = (const float*)d_in[0];   // (16,16,128,128,3,3)
    const float* g = (const float*)d_in[1];   // (16,16,8,3)
    float* out = (float*)d_out;               // (16,16,130,130,3,3)
    float* Gws = (float*)d_ws;                // 20736 floats: G[o][i][3][3][3][3]
    float* Wm  = Gws + OCH*CIN*9*9;           // 6912 floats: Wm[144][48]

    k_build_G <<<   9, 256, 0, stream>>>(g, Gws);          // 2304 threads
    k_build_Wm<<<  27, 256, 0, stream>>>(Gws, Wm);         // 6912 threads
    k_interior<<<4096, 128, 0, stream>>>(x, Wm, out);      // 16384 waves
    k_border  <<< 516, 256, 0, stream>>>(x, Gws, out);     // 132096 threads
}